// HierarchicalLoss_75642964017719
// MI455X (gfx1250) — compile-verified
//
#include <hip/hip_runtime.h>
#include <stdint.h>

// ---------------------------------------------------------------------------
// Hierarchical CE loss, MI455X (gfx1250). Memory-bound: 328MB @ 23.3TB/s
// ~= 14us floor. Strategy:
//   * TDM (tensor_load_to_lds) DMAs 40KB rows into LDS, double-buffered:
//     8 rows per workgroup, prefetch row r+1 while computing row r
//     (TENSORcnt is in-order per wave -> s_wait_tensorcnt 1 == row r ready).
//   * Single global row max m; all three prefix LSEs share it:
//       lse_l = m + ln( sum_{j<S_l} exp(x_j - m) )   (S_l = 500/2000/10000)
//     -> exactly one v_exp_f32 per element, one max reduction per row.
//   * float4 LDS traffic (ds_load_b128), fused 3-way sum reduction.
//   * Final 8192-element mean uses V_WMMA_F32_16X16X4_F32 with A=ones:
//     D[m,n] = sum_k B[k,n] + C[m,n], so each WMMA folds 64 values into a
//     chained accumulator (exact, deterministic, off the critical path).
// ---------------------------------------------------------------------------

#define B_ROWS   8192
#define C_COLS   10000
#define C_VEC    (C_COLS / 4)     // 2500 float4 per row
#define L1_VEC   (2000 / 4)       // 500
#define L2_VEC   (500 / 4)        // 125
#define RPB      8                // rows per block (pipelined)
#define NBLOCKS  (B_ROWS / RPB)   // 1024
#define NTHREADS 256
#define NWAVES   (NTHREADS / 32)
#define LOG2E_F  1.4426950408889634f
#define LN2_F    0.6931471805599453f

typedef uint32_t u32x4 __attribute__((ext_vector_type(4)));
typedef int      i32x8 __attribute__((ext_vector_type(8)));
typedef int      i32x4 __attribute__((ext_vector_type(4)));
typedef float    v2f   __attribute__((ext_vector_type(2)));
typedef float    v8f   __attribute__((ext_vector_type(8)));

// Issue one TDM descriptor: contiguous 40000B row  global -> LDS.
__device__ __forceinline__ void issue_tdm_row(const float* gsrc, uint32_t lds_off) {
    const uint64_t ga = (uint64_t)(uintptr_t)gsrc;
    u32x4 g0;
    g0[0] = 1u;                                       // count=1, user mode
    g0[1] = lds_off;                                  // lds_addr (bytes)
    g0[2] = (uint32_t)ga;                             // global_addr[31:0]
    g0[3] = (uint32_t)((ga >> 32) & 0x1FFFFFFu)       // global_addr[56:32]
          | (2u << 30);                               // type=2 ("image")
    i32x8 g1;
    g1[0] = (int)(2u << 16);                          // data_size=4B
    g1[1] = (int)(((uint32_t)C_COLS & 0xFFFFu) << 16);   // tensor_dim0 lo16
    g1[2] = (int)(((uint32_t)C_COLS >> 16) | (1u << 16));// dim0 hi | tensor_dim1=1
    g1[3] = (int)(((uint32_t)C_COLS & 0xFFFFu) << 16);   // tile_dim0 = 10000
    g1[4] = 1;                                        // tile_dim1=1
    g1[5] = C_COLS;                                   // tensor_dim0_stride lo32
    g1[6] = 0;
    g1[7] = 0;
    i32x4 g2 = {0, 0, 0, 0};
    i32x4 g3 = {0, 0, 0, 0};
#if defined(__clang_major__) && (__clang_major__ >= 23)
    i32x8 g4 = {0, 0, 0, 0, 0, 0, 0, 0};
    __builtin_amdgcn_tensor_load_to_lds(g0, g1, g2, g3, g4, 0);
#else
    __builtin_amdgcn_tensor_load_to_lds(g0, g1, g2, g3, 0);
#endif
}

// Block-wide max over 256 threads; result returned to all threads.
__device__ __forceinline__ float block_reduce_max(float v, float* scr, int tid) {
    #pragma unroll
    for (int o = 16; o > 0; o >>= 1) v = fmaxf(v, __shfl_xor(v, o, 32));
    const int wid = tid >> 5, lane = tid & 31;
    if (lane == 0) scr[wid] = v;
    __syncthreads();
    if (wid == 0) {
        float w = (lane < NWAVES) ? scr[lane] : -INFINITY;
        #pragma unroll
        for (int o = 16; o > 0; o >>= 1) w = fmaxf(w, __shfl_xor(w, o, 32));
        if (lane == 0) scr[0] = w;
    }
    __syncthreads();
    float r = scr[0];
    __syncthreads();                 // protect scr before reuse
    return r;
}

// Fused block-wide sum of three accumulators (one LDS round-trip).
__device__ __forceinline__ void block_reduce_sum3(float& s0, float& s1, float& s2,
                                                  float* scr, int tid) {
    #pragma unroll
    for (int o = 16; o > 0; o >>= 1) {
        s0 += __shfl_xor(s0, o, 32);
        s1 += __shfl_xor(s1, o, 32);
        s2 += __shfl_xor(s2, o, 32);
    }
    const int wid = tid >> 5, lane = tid & 31;
    if (lane == 0) { scr[wid] = s0; scr[NWAVES + wid] = s1; scr[2 * NWAVES + wid] = s2; }
    __syncthreads();
    if (wid == 0) {
        float a = (lane < NWAVES) ? scr[lane]              : 0.0f;
        float b = (lane < NWAVES) ? scr[NWAVES + lane]     : 0.0f;
        float c = (lane < NWAVES) ? scr[2 * NWAVES + lane] : 0.0f;
        #pragma unroll
        for (int o = 16; o > 0; o >>= 1) {
            a += __shfl_xor(a, o, 32);
            b += __shfl_xor(b, o, 32);
            c += __shfl_xor(c, o, 32);
        }
        if (lane == 0) { scr[0] = a; scr[NWAVES] = b; scr[2 * NWAVES] = c; }
    }
    __syncthreads();
    s0 = scr[0]; s1 = scr[NWAVES]; s2 = scr[2 * NWAVES];
    // next write to scr happens after the caller's end-of-row barrier
}

__global__ __launch_bounds__(NTHREADS)
void hier_ce_rows(const float* __restrict__ logits,
                  const int*   __restrict__ targets,
                  float*       __restrict__ row_loss) {
    __shared__ float4 sbuf[2][C_VEC];        // 2 x 40000B row buffers
    __shared__ float  sred[3 * NWAVES];

    const int tid  = threadIdx.x;
    const int base = blockIdx.x * RPB;

    // Prime the pipeline: DMA row 0 into buffer 0.
    if (tid < 32) {
        issue_tdm_row(logits + (size_t)base * C_COLS,
                      (uint32_t)(uintptr_t)&sbuf[0][0]);
    }

    for (int r = 0; r < RPB; ++r) {
        // Prefetch next row into the alternate buffer, then wait for row r.
        // TDM ops complete in order per wave, so tensorcnt<=1 => row r landed.
        if (tid < 32) {
            if (r + 1 < RPB) {
                issue_tdm_row(logits + (size_t)(base + r + 1) * C_COLS,
                              (uint32_t)(uintptr_t)&sbuf[(r + 1) & 1][0]);
                __builtin_amdgcn_s_wait_tensorcnt(1);
            } else {
                __builtin_amdgcn_s_wait_tensorcnt(0);
            }
        }
        __syncthreads();                         // row r visible to all waves

        const float4* buf = sbuf[r & 1];

        // Pass 1: global row max (shared by all three prefix LSEs).
        float m = -INFINITY;
        for (int c = tid; c < C_VEC; c += NTHREADS) {
            float4 v = buf[c];
            m = fmaxf(m, fmaxf(fmaxf(v.x, v.y), fmaxf(v.z, v.w)));
        }
        m = block_reduce_max(m, sred, tid);

        // Pass 2: prefix sums of e = exp(x - m) over [0,500),[0,2000),[0,10000).
        float s0 = 0.0f, s1 = 0.0f, s2 = 0.0f;
        for (int c = tid; c < C_VEC; c += NTHREADS) {
            float4 v = buf[c];
            float e = exp2f((v.x - m) * LOG2E_F)
                    + exp2f((v.y - m) * LOG2E_F)
                    + exp2f((v.z - m) * LOG2E_F)
                    + exp2f((v.w - m) * LOG2E_F);
            s0 += e;
            if (c < L1_VEC) s1 += e;
            if (c < L2_VEC) s2 += e;
        }
        block_reduce_sum3(s0, s1, s2, sred, tid);

        if (tid == 0) {
            const int   t  = targets[base + r];               // t in [0,500)
            const float xt = ((const float*)buf)[t];          // logits[row, t]
            const float lse0 = m + log2f(s0) * LN2_F;
            const float lse1 = m + log2f(s1) * LN2_F;
            const float lse2 = m + log2f(s2) * LN2_F;
            row_loss[base + r] = (lse0 + lse1 + lse2) - 3.0f * xt;
        }
        __syncthreads();   // everyone done with buf[r&1] before it is re-DMAed
    }
}

// Deterministic final mean over the 8192 per-row partials, accumulated on the
// matrix pipe: D = ones(16x4) x B(4x16) + C  =>  each WMMA folds this wave's
// next 64 values into a 16-lane running accumulator (row 0 of C).
__global__ __launch_bounds__(NTHREADS)
void final_reduce(const float* __restrict__ row_loss, float* __restrict__ out) {
    __shared__ float sred[NWAVES];
    const int tid  = threadIdx.x;
    const int wid  = tid >> 5;
    const int lane = tid & 31;

    // Wave w owns elements [w*1024, (w+1)*1024): 16 iterations x 64 values.
    const int ELEMS_PER_WAVE = B_ROWS / NWAVES;          // 1024
    const int ITERS          = ELEMS_PER_WAVE / 64;      // 16

    const v2f* p = (const v2f*)row_loss;                 // float2-granular
    const int  wave_base2 = wid * (ELEMS_PER_WAVE / 2);  // in float2 units

    v2f a; a[0] = 1.0f; a[1] = 1.0f;                     // A = ones 16x4
    v8f c = {};                                          // running accumulator

    #pragma unroll
    for (int i = 0; i < ITERS; ++i) {
        // Lane l supplies B VGPR0/VGPR1 = 2 consecutive floats; mapping of
        // lanes to B[k][n] is irrelevant: sum_n D[0,n] == sum of all 64.
        v2f b = p[wave_base2 + i * 32 + lane];
        c = __builtin_amdgcn_wmma_f32_16x16x4_f32(
                /*neg_a=*/false, a, /*neg_b=*/false, b,
                /*c_mod=*/(short)0, c, /*reuse_a=*/false, /*reuse_b=*/false);
    }

    // D row 0 lives in VGPR0, lanes 0..15 (lanes 16..31 duplicate row 8).
    float w = c[0];
    #pragma unroll
    for (int o = 8; o > 0; o >>= 1) w += __shfl_xor(w, o, 16);  // fold lanes 0..15

    if (lane == 0) sred[wid] = w;
    __syncthreads();
    if (tid == 0) {
        float t = 0.0f;
        #pragma unroll
        for (int v = 0; v < NWAVES; ++v) t += sred[v];
        out[0] = t * (1.0f / (float)B_ROWS);
    }
}

extern "C" void kernel_launch(void* const* d_in, const int* in_sizes, int n_in,
                              void* d_out, int out_size, void* d_ws, size_t ws_size,
                              hipStream_t stream) {
    const float* logits  = (const float*)d_in[0];
    const int*   targets = (const int*)d_in[1];
    // d_in[2..4] (level permutations) are provably irrelevant: LSE over a
    // permuted index set == LSE over the set, and idx[inv[t]] == t.
    (void)in_sizes; (void)n_in; (void)out_size; (void)ws_size;

    float* row_loss = (float*)d_ws;   // 8192 floats of scratch

    hier_ce_rows<<<NBLOCKS, NTHREADS, 0, stream>>>(logits, targets, row_loss);
    final_reduce<<<1, NTHREADS, 0, stream>>>(row_loss, (float*)d_out);
}